// Net_59725815218768
// MI455X (gfx1250) — compile-verified
//
#include <hip/hip_runtime.h>
#include <cstddef>
#include <cstdint>

// ---------------- problem constants (from reference) ----------------
#define DIN     512     // input feature dim
#define LAYERS  8       // parallel GATConv heads in stage 1
#define HDIM    8       // per-layer output dim
#define F1      64      // LAYERS*HDIM, fused stage-1 GEMM width
#define CDIM    16      // stage-2 output classes
#define NEG_SLOPE 0.2f

typedef float v2f __attribute__((ext_vector_type(2)));
typedef float v8f __attribute__((ext_vector_type(8)));

// ---------------- CDNA5 async-copy helpers ----------------
#if __has_builtin(__builtin_amdgcn_s_wait_asynccnt)
#define WAIT_ASYNCCNT(n) __builtin_amdgcn_s_wait_asynccnt(n)
#else
#define WAIT_ASYNCCNT(n) asm volatile("s_wait_asynccnt %0" :: "n"(n) : "memory")
#endif

// global -> LDS direct async copy (16B per lane), tracked by ASYNCcnt
__device__ __forceinline__ void async_copy_b128(unsigned lds_off, const void* gaddr) {
    asm volatile("global_load_async_to_lds_b128 %0, %1, off"
                 :: "v"(lds_off), "v"(gaddr)
                 : "memory");
}
__device__ __forceinline__ unsigned lds_offset(const void* p) {
    // generic pointer to an LDS object: low 32 bits = group-segment byte offset
    return (unsigned)(size_t)p;
}

// ---------------- helpers ----------------
__device__ __forceinline__ float leaky(float f) {
    return f >= 0.f ? f : NEG_SLOPE * f;
}
// order-preserving float<->uint mapping so atomicMax(unsigned) == float max
__device__ __forceinline__ unsigned enc_f32(float f) {
    unsigned b = __float_as_uint(f);
    return (b & 0x80000000u) ? ~b : (b | 0x80000000u);
}
__device__ __forceinline__ float dec_f32(unsigned u) {
    return (u & 0x80000000u) ? __uint_as_float(u & 0x7fffffffu)
                             : __uint_as_float(~u);
}

// ---------------- kernel 0: repack W_hidden[8][512][8] -> W_all[512][64] ----------------
__global__ void k_repack(const float* __restrict__ Wh, float* __restrict__ Wall, int Din) {
    int idx = blockIdx.x * blockDim.x + threadIdx.x;
    if (idx >= Din * F1) return;
    int d = idx >> 6;          // 0..511
    int t = idx & 63;          // 0..63
    int i = t >> 3;            // layer
    int j = t & 7;             // comp
    Wall[idx] = Wh[(i * Din + d) * HDIM + j];
}

// ---------------- kernel 1: Hfeat[N][64] = x[N][512] @ W_all[512][64] ----------------
// f32 WMMA, double-buffered async global->LDS staging (ASYNCcnt ping-pong).
__global__ __launch_bounds__(256) void k_gemm1(const float* __restrict__ X,
                                               const float* __restrict__ Wall,
                                               float* __restrict__ Hf, int M) {
    __shared__ float As[2][32][68];   // 32 rows x 64 k-vals, pad 68 -> conflict-free A reads
    __shared__ float Bs[2][64][72];   // 64 k-vals x 64 cols, pad 72 -> disjoint half-wave banks

    const int tid   = threadIdx.x;
    const int mBase = blockIdx.x * 32;
    if (mBase >= M) return;
    const int wave = tid >> 5;
    const int lane = tid & 31;
    const int l16  = lane & 15;
    const int half = lane >> 4;
    const int r0   = (wave >> 2) << 4;   // 0 or 16 (row tile in block)
    const int c0   = (wave & 3) << 4;    // 0/16/32/48 (col tile)

    // each thread copies exactly 6 x 16B per K-chunk: 2 for A, 4 for B
    auto issue_chunk = [&](int buf, int k0) {
        #pragma unroll
        for (int it = 0; it < 2; ++it) {
            int idx = tid + it * 256;             // 0..511 over 32x16 float4 tile
            int row = idx >> 4, q = idx & 15;
            async_copy_b128(lds_offset(&As[buf][row][q * 4]),
                            X + (size_t)(mBase + row) * DIN + k0 + q * 4);
        }
        #pragma unroll
        for (int it = 0; it < 4; ++it) {
            int idx = tid + it * 256;             // 0..1023 over 64x16 float4 tile
            int row = idx >> 4, q = idx & 15;
            async_copy_b128(lds_offset(&Bs[buf][row][q * 4]),
                            Wall + (size_t)(k0 + row) * F1 + q * 4);
        }
    };

    v8f acc = {0.f, 0.f, 0.f, 0.f, 0.f, 0.f, 0.f, 0.f};

    const int NITER = DIN / 64;                   // 8 K-chunks
    issue_chunk(0, 0);                            // prologue: fill buffer 0
    #pragma unroll 2
    for (int i = 0; i < NITER; ++i) {
        const int b = i & 1;
        if (i + 1 < NITER) {
            issue_chunk(1 - b, (i + 1) * 64);     // prefetch next chunk (safe: 1-b idle since i-1's barrier)
            WAIT_ASYNCCNT(6);                     // older 6 copies (buffer b) done; newer 6 in flight
        } else {
            WAIT_ASYNCCNT(0);
        }
        __syncthreads();                          // all waves' chunk-b data visible in LDS
        // 16 WMMA steps of K=4 each
        #pragma unroll
        for (int kk = 0; kk < 64; kk += 4) {
            v2f a, bb;
            a.x  = As[b][r0 + l16][kk + 2 * half];
            a.y  = As[b][r0 + l16][kk + 2 * half + 1];
            bb.x = Bs[b][kk + 2 * half][c0 + l16];
            bb.y = Bs[b][kk + 2 * half + 1][c0 + l16];
            acc = __builtin_amdgcn_wmma_f32_16x16x4_f32(
                false, a, false, bb, (short)0, acc, false, false);
        }
        __syncthreads();                          // buffer b free for iteration i+2's prefetch
    }
    // C/D layout: VGPR r -> row r (lanes 0-15) / row r+8 (lanes 16-31), col = l16
    #pragma unroll
    for (int r = 0; r < 8; ++r)
        Hf[(size_t)(mBase + r0 + r + 8 * half) * F1 + c0 + l16] = acc[r];
}

// ---------------- kernel 2: attention logits per node/layer ----------------
__global__ void k_alpha1(const float* __restrict__ Hf,
                         const float* __restrict__ aS, const float* __restrict__ aD,
                         float* __restrict__ alS, float* __restrict__ alD, int N) {
    int idx = blockIdx.x * blockDim.x + threadIdx.x;
    if (idx >= N * LAYERS) return;
    int n = idx >> 3, l = idx & 7;
    const float* h = Hf + (size_t)n * F1 + l * HDIM;
    float ss = 0.f, dd = 0.f;
    #pragma unroll
    for (int j = 0; j < HDIM; ++j) {
        ss += h[j] * aS[l * HDIM + j];
        dd += h[j] * aD[l * HDIM + j];
    }
    alS[idx] = ss;
    alD[idx] = dd;
}

// ---------------- kernel 3: per-dst segment max (stage 1) ----------------
__global__ void k_edgemax1(const int* __restrict__ ei, int E, int N,
                           const float* __restrict__ alS, const float* __restrict__ alD,
                           unsigned* __restrict__ m) {
    int idx = blockIdx.x * blockDim.x + threadIdx.x;
    if (idx >= (E + N) * LAYERS) return;
    int e = idx >> 3, l = idx & 7;
    int s, d;
    if (e < E) { s = ei[e]; d = ei[E + e]; } else { s = d = e - E; }
    float v = leaky(alS[s * LAYERS + l] + alD[d * LAYERS + l]);
    atomicMax(&m[d * LAYERS + l], enc_f32(v));
}

// ---------------- kernel 4: fused exp + weighted scatter-add (stage 1) ----------------
// 64 threads per edge: thread covers one (layer,comp); j==0 thread also adds the denom.
__global__ __launch_bounds__(256) void k_edgeacc1(const int* __restrict__ ei, int E, int N,
                                                  const float* __restrict__ alS,
                                                  const float* __restrict__ alD,
                                                  const unsigned* __restrict__ m,
                                                  const float* __restrict__ Hf,
                                                  float* __restrict__ sbuf,
                                                  float* __restrict__ accb) {
    int t = threadIdx.x;
    int e = blockIdx.x * 4 + (t >> 6);
    if (e >= E + N) return;
    int comp = t & 63, l = comp >> 3, j = comp & 7;
    int s, d;
    if (e < E) { s = ei[e]; d = ei[E + e]; } else { s = d = e - E; }
    float mv = dec_f32(m[d * LAYERS + l]);
    float ex = expf(leaky(alS[s * LAYERS + l] + alD[d * LAYERS + l]) - mv);
    atomicAdd(&accb[(size_t)d * F1 + comp], ex * Hf[(size_t)s * F1 + comp]);
    if (j == 0) atomicAdd(&sbuf[d * LAYERS + l], ex);
}

// ---------------- kernel 5: normalize + bias + layer-average + ELU ----------------
__global__ void k_node1(const float* __restrict__ accb, const float* __restrict__ sbuf,
                        const float* __restrict__ bias, float* __restrict__ h1, int N) {
    int n = blockIdx.x * blockDim.x + threadIdx.x;
    if (n >= N) return;
    float inv[LAYERS];
    #pragma unroll
    for (int l = 0; l < LAYERS; ++l) inv[l] = 1.f / sbuf[n * LAYERS + l];
    #pragma unroll
    for (int j = 0; j < HDIM; ++j) {
        float t = 0.f;
        #pragma unroll
        for (int l = 0; l < LAYERS; ++l)
            t += accb[(size_t)n * F1 + l * HDIM + j] * inv[l] + bias[l * HDIM + j];
        t *= 0.125f;                              // average of 8 layers
        h1[n * HDIM + j] = t > 0.f ? t : expf(t) - 1.f;   // ELU
    }
}

// ---------------- kernel 6: h2[N][16] = h1[N][8] @ W_out[8][16] (f32 WMMA) ----------------
__global__ __launch_bounds__(256) void k_gemm2(const float* __restrict__ h1,
                                               const float* __restrict__ Wo,
                                               float* __restrict__ h2, int tiles) {
    int wave = threadIdx.x >> 5;
    int gw = blockIdx.x * 8 + wave;
    if (gw >= tiles) return;                     // wave-uniform: EXEC stays all-ones
    int lane = threadIdx.x & 31, l16 = lane & 15, half = lane >> 4;
    int m0 = gw * 16;

    v8f acc = {0.f, 0.f, 0.f, 0.f, 0.f, 0.f, 0.f, 0.f};
    #pragma unroll
    for (int k = 0; k < HDIM; k += 4) {
        v2f a, b;
        a.x = h1[(size_t)(m0 + l16) * HDIM + k + 2 * half];
        a.y = h1[(size_t)(m0 + l16) * HDIM + k + 2 * half + 1];
        b.x = Wo[(k + 2 * half) * CDIM + l16];
        b.y = Wo[(k + 2 * half + 1) * CDIM + l16];
        acc = __builtin_amdgcn_wmma_f32_16x16x4_f32(
            false, a, false, b, (short)0, acc, false, false);
    }
    #pragma unroll
    for (int r = 0; r < 8; ++r)
        h2[(size_t)(m0 + r + 8 * half) * CDIM + l16] = acc[r];
}

// ---------------- kernel 7: stage-2 attention logits ----------------
__global__ void k_alpha2(const float* __restrict__ h2,
                         const float* __restrict__ aS, const float* __restrict__ aD,
                         float* __restrict__ alS, float* __restrict__ alD, int N) {
    int n = blockIdx.x * blockDim.x + threadIdx.x;
    if (n >= N) return;
    const float* h = h2 + (size_t)n * CDIM;
    float ss = 0.f, dd = 0.f;
    #pragma unroll
    for (int c = 0; c < CDIM; ++c) { ss += h[c] * aS[c]; dd += h[c] * aD[c]; }
    alS[n] = ss;
    alD[n] = dd;
}

// ---------------- kernel 8: stage-2 segment max ----------------
__global__ void k_edgemax2(const int* __restrict__ ei, int E, int N,
                           const float* __restrict__ alS, const float* __restrict__ alD,
                           unsigned* __restrict__ m2) {
    int e = blockIdx.x * blockDim.x + threadIdx.x;
    if (e >= E + N) return;
    int s, d;
    if (e < E) { s = ei[e]; d = ei[E + e]; } else { s = d = e - E; }
    atomicMax(&m2[d], enc_f32(leaky(alS[s] + alD[d])));
}

// ---------------- kernel 9: stage-2 fused exp + weighted scatter-add ----------------
__global__ __launch_bounds__(256) void k_edgeacc2(const int* __restrict__ ei, int E, int N,
                                                  const float* __restrict__ alS,
                                                  const float* __restrict__ alD,
                                                  const unsigned* __restrict__ m2,
                                                  const float* __restrict__ h2,
                                                  float* __restrict__ s2,
                                                  float* __restrict__ acc2) {
    int t = threadIdx.x;
    int e = blockIdx.x * 16 + (t >> 4);
    if (e >= E + N) return;
    int c = t & 15;
    int s, d;
    if (e < E) { s = ei[e]; d = ei[E + e]; } else { s = d = e - E; }
    float ex = expf(leaky(alS[s] + alD[d]) - dec_f32(m2[d]));
    atomicAdd(&acc2[(size_t)d * CDIM + c], ex * h2[(size_t)s * CDIM + c]);
    if (c == 0) atomicAdd(&s2[d], ex);
}

// ---------------- kernel 10: normalize + bias + log_softmax ----------------
__global__ void k_final(const float* __restrict__ acc2, const float* __restrict__ s2,
                        const float* __restrict__ bias, float* __restrict__ out, int N) {
    int n = blockIdx.x * blockDim.x + threadIdx.x;
    if (n >= N) return;
    float inv = 1.f / s2[n];
    float o[CDIM], mx = -3.402823466e+38f;
    #pragma unroll
    for (int c = 0; c < CDIM; ++c) {
        o[c] = acc2[(size_t)n * CDIM + c] * inv + bias[c];
        mx = fmaxf(mx, o[c]);
    }
    float sum = 0.f;
    #pragma unroll
    for (int c = 0; c < CDIM; ++c) sum += expf(o[c] - mx);
    float lse = logf(sum) + mx;
    #pragma unroll
    for (int c = 0; c < CDIM; ++c) out[(size_t)n * CDIM + c] = o[c] - lse;
}

// ---------------- host launcher ----------------
extern "C" void kernel_launch(void* const* d_in, const int* in_sizes, int n_in,
                              void* d_out, int out_size, void* d_ws, size_t ws_size,
                              hipStream_t stream) {
    (void)n_in; (void)out_size; (void)ws_size;
    const float* x   = (const float*)d_in[0];
    const int*   ei  = (const int*)d_in[1];     // edge_index [2][E], harness delivers int32
    const float* Wh  = (const float*)d_in[2];   // [8][512][8]
    const float* aSh = (const float*)d_in[3];   // [8][8]
    const float* aDh = (const float*)d_in[4];   // [8][8]
    const float* bh  = (const float*)d_in[5];   // [8][8]
    const float* Wo  = (const float*)d_in[6];   // [8][16]
    const float* aSo = (const float*)d_in[7];   // [16]
    const float* aDo = (const float*)d_in[8];   // [16]
    const float* bo  = (const float*)d_in[9];   // [16]
    float* out = (float*)d_out;

    const int N  = in_sizes[0] / DIN;           // 100000
    const int E  = in_sizes[1] / 2;             // 1600000
    const int EA = E + N;                       // with self-loops

    // workspace layout (256B aligned slabs)
    char* ws = (char*)d_ws;
    size_t off = 0;
    auto alloc = [&](size_t bytes) -> size_t {
        size_t o = (off + 255) & ~(size_t)255;
        off = o + bytes;
        return o;
    };
    size_t oWall  = alloc((size_t)DIN * F1 * 4);
    size_t oHf    = alloc((size_t)N * F1 * 4);
    size_t oAlS   = alloc((size_t)N * LAYERS * 4);
    size_t oAlD   = alloc((size_t)N * LAYERS * 4);
    size_t oH1    = alloc((size_t)N * HDIM * 4);
    size_t oH2    = alloc((size_t)N * CDIM * 4);
    size_t oAlS2  = alloc((size_t)N * 4);
    size_t oAlD2  = alloc((size_t)N * 4);
    // zero-initialized region (one memset): m, s, acc, m2, s2, acc2
    size_t oZero  = (off + 255) & ~(size_t)255;
    off = oZero;
    size_t oM     = alloc((size_t)N * LAYERS * 4);
    size_t oS     = alloc((size_t)N * LAYERS * 4);
    size_t oAcc   = alloc((size_t)N * F1 * 4);
    size_t oM2    = alloc((size_t)N * 4);
    size_t oS2    = alloc((size_t)N * 4);
    size_t oAcc2  = alloc((size_t)N * CDIM * 4);
    size_t zeroBytes = off - oZero;

    float*    Wall = (float*)(ws + oWall);
    float*    Hf   = (float*)(ws + oHf);
    float*    alS  = (float*)(ws + oAlS);
    float*    alD  = (float*)(ws + oAlD);
    float*    h1   = (float*)(ws + oH1);
    float*    h2   = (float*)(ws + oH2);
    float*    alS2 = (float*)(ws + oAlS2);
    float*    alD2 = (float*)(ws + oAlD2);
    unsigned* mBuf = (unsigned*)(ws + oM);
    float*    sBuf = (float*)(ws + oS);
    float*    acc  = (float*)(ws + oAcc);
    unsigned* m2   = (unsigned*)(ws + oM2);
    float*    s2   = (float*)(ws + oS2);
    float*    acc2 = (float*)(ws + oAcc2);

    const int B = 256;
    hipMemsetAsync(ws + oZero, 0, zeroBytes, stream);   // graph-capturable memset node

    k_repack <<<(DIN * F1 + B - 1) / B, B, 0, stream>>>(Wh, Wall, DIN);
    k_gemm1  <<<(N + 31) / 32, B, 0, stream>>>(x, Wall, Hf, N);
    k_alpha1 <<<(N * LAYERS + B - 1) / B, B, 0, stream>>>(Hf, aSh, aDh, alS, alD, N);
    k_edgemax1<<<((size_t)EA * LAYERS + B - 1) / B, B, 0, stream>>>(ei, E, N, alS, alD, mBuf);
    k_edgeacc1<<<(EA + 3) / 4, B, 0, stream>>>(ei, E, N, alS, alD, mBuf, Hf, sBuf, acc);
    k_node1  <<<(N + B - 1) / B, B, 0, stream>>>(acc, sBuf, bh, h1, N);
    k_gemm2  <<<((N / 16) + 7) / 8, B, 0, stream>>>(h1, Wo, h2, N / 16);
    k_alpha2 <<<(N + B - 1) / B, B, 0, stream>>>(h2, aSo, aDo, alS2, alD2, N);
    k_edgemax2<<<(EA + B - 1) / B, B, 0, stream>>>(ei, E, N, alS2, alD2, m2);
    k_edgeacc2<<<(EA + 15) / 16, B, 0, stream>>>(ei, E, N, alS2, alD2, m2, h2, s2, acc2);
    k_final  <<<(N + B - 1) / B, B, 0, stream>>>(acc2, s2, bo, out, N);
}